// ELBO_60447369724418
// MI455X (gfx1250) — compile-verified
//
#include <hip/hip_runtime.h>
#include <cstdint>
#include <cmath>

#define NTOT 5120
#define BSZ  32
#define TMAXC 20
#define PIXC 1024
#define DBLK 160
#define NB   128
#define KCH  32                 // K staging chunk for SYRK
#define LDK  36                 // padded LDS row stride (conflict-free)
#define LOG_2PI 1.8378770664093453f

typedef __attribute__((ext_vector_type(2))) float v2f;
typedef __attribute__((ext_vector_type(8))) float v8f;

// ---------------- elementwise / small kernels ----------------

__global__ void k_zero(float* p, int n) {
  int i = blockIdx.x * blockDim.x + threadIdx.x;
  if (i < n) p[i] = 0.0f;
}

__global__ void k_add4(const float4* __restrict__ a, const float4* __restrict__ b,
                       float4* __restrict__ w, size_t n4) {
  size_t i = (size_t)blockIdx.x * blockDim.x + threadIdx.x;
  size_t stride = (size_t)gridDim.x * blockDim.x;
  for (; i < n4; i += stride) {
    float4 x = a[i], y = b[i];
    w[i] = make_float4(x.x + y.x, x.y + y.y, x.z + y.z, x.w + y.w);
  }
}

__global__ void k_vsub(const float* __restrict__ a, const float* __restrict__ b,
                       float* __restrict__ o, int n) {
  int i = blockIdx.x * blockDim.x + threadIdx.x;
  if (i < n) o[i] = a[i] - b[i];
}

__global__ void k_diag(const float* __restrict__ K, float* __restrict__ d, int n) {
  int i = blockIdx.x * blockDim.x + threadIdx.x;
  if (i < n) d[i] = K[(size_t)i * NTOT + i];
}

// ---------------- blocked Cholesky: panel factor ----------------

__global__ void k_potrf(float* __restrict__ W, int pcol) {
  extern __shared__ float Ls[];              // NB x (NB+1)
  const int LD = NB + 1;
  const int tid = threadIdx.x;               // NB threads, thread == row
  for (int c = 0; c < NB; ++c)
    Ls[tid * LD + c] = W[(size_t)(pcol + tid) * NTOT + pcol + c];
  __syncthreads();
  for (int k = 0; k < NB; ++k) {
    if (tid == k) Ls[k * LD + k] = sqrtf(Ls[k * LD + k]);
    __syncthreads();
    if (tid > k) Ls[tid * LD + k] /= Ls[k * LD + k];
    __syncthreads();
    if (tid > k) {
      float lik = Ls[tid * LD + k];
      for (int j = k + 1; j <= tid; ++j)
        Ls[tid * LD + j] -= lik * Ls[j * LD + k];
    }
    __syncthreads();
  }
  for (int c = 0; c <= tid; ++c)
    W[(size_t)(pcol + tid) * NTOT + pcol + c] = Ls[tid * LD + c];
}

// ---------------- panel TRSM: X * L11^T = A21 ----------------

__global__ void k_trsm(float* __restrict__ W, int pcol, int srow) {
  extern __shared__ float lds[];
  float* L11 = lds;                          // NB x (NB+1)
  float* Rw  = lds + NB * (NB + 1);          // NB x (NB+1), one row per thread
  const int LD = NB + 1;
  const int tid = threadIdx.x;               // NB threads
  const int row = srow + blockIdx.x * NB + tid;
  for (int c = 0; c < NB; ++c)
    L11[tid * LD + c] = W[(size_t)(pcol + tid) * NTOT + pcol + c];
  for (int c = 0; c < NB; ++c)
    Rw[tid * LD + c] = W[(size_t)row * NTOT + pcol + c];
  __syncthreads();
  // forward substitution along the row; rows are independent (no barriers needed)
  for (int c = 0; c < NB; ++c) {
    float v = Rw[tid * LD + c];
    for (int t = 0; t < c; ++t)
      v -= Rw[tid * LD + t] * L11[c * LD + t];    // L11[c][*] broadcast read
    Rw[tid * LD + c] = v / L11[c * LD + c];
  }
  for (int c = 0; c < NB; ++c)
    W[(size_t)row * NTOT + pcol + c] = Rw[tid * LD + c];
}

// ---------------- trailing SYRK via V_WMMA_F32_16X16X4_F32 ----------------
// 128x128 output tile per 256-thread (8-wave) workgroup; each wave owns a
// 32x64 quadrant = 2x4 f32 accumulators (64 acc VGPRs -> no spills). K staged
// through LDS in 32-wide chunks: 6 fragment loads feed 8 WMMAs per k-step.

__global__ void __launch_bounds__(256) k_syrk(float* __restrict__ W, int pcol, int s) {
  const int bx = blockIdx.x, by = blockIdx.y;
  if (by > bx) return;                        // lower-triangular tiles only
  extern __shared__ float lds[];
  float* As = lds;                            // 128 x LDK
  float* Bs = lds + 128 * LDK;                // 128 x LDK
  const int rowbase = s + bx * 128;
  const int colbase = s + by * 128;
  const int tid  = threadIdx.x;               // 256 threads
  const int wave = tid >> 5;                  // 0..7, arranged 4 (rows) x 2 (cols)
  const int lane = tid & 31;
  const int lo = lane & 15;                   // M/N index within 16
  const int hi = lane >> 4;                   // selects K pair {0,1} vs {2,3}
  const int rowOff = (wave >> 1) * 32;
  const int colOff = (wave & 1) * 64;

  v8f acc[2][4] = {};

  for (int kc = 0; kc < NB; kc += KCH) {
    __syncthreads();
    // stage 128 rows x 32 k of both panels: 1024 float4 per array, 4/thread
    for (int it = 0; it < 4; ++it) {
      int idx = tid + it * 256;
      int r  = idx >> 3;                      // row 0..127
      int c4 = idx & 7;                       // float4 column 0..7 (k 0..31)
      float4 va = *(const float4*)&W[(size_t)(rowbase + r) * NTOT + pcol + kc + c4 * 4];
      float* da = &As[r * LDK + c4 * 4];
      da[0] = va.x; da[1] = va.y; da[2] = va.z; da[3] = va.w;
      float4 vb = *(const float4*)&W[(size_t)(colbase + r) * NTOT + pcol + kc + c4 * 4];
      float* db = &Bs[r * LDK + c4 * 4];
      db[0] = vb.x; db[1] = vb.y; db[2] = vb.z; db[3] = vb.w;
    }
    __syncthreads();

    #pragma unroll
    for (int kk = 0; kk < KCH; kk += 4) {
      const int ks = kk + 2 * hi;
      v2f af[2], bf[4];
      #pragma unroll
      for (int i = 0; i < 2; ++i) {
        af[i].x = As[(rowOff + 16 * i + lo) * LDK + ks];
        af[i].y = As[(rowOff + 16 * i + lo) * LDK + ks + 1];
      }
      #pragma unroll
      for (int j = 0; j < 4; ++j) {
        bf[j].x = Bs[(colOff + 16 * j + lo) * LDK + ks];
        bf[j].y = Bs[(colOff + 16 * j + lo) * LDK + ks + 1];
      }
      #pragma unroll
      for (int i = 0; i < 2; ++i)
        #pragma unroll
        for (int j = 0; j < 4; ++j)
          acc[i][j] = __builtin_amdgcn_wmma_f32_16x16x4_f32(
              false, af[i], false, bf[j], (short)0, acc[i][j], false, false);
    }
  }

  // C/D layout: VGPR r -> row = r + 8*hi, col = lo (per 16x16 fragment)
  #pragma unroll
  for (int i = 0; i < 2; ++i)
    #pragma unroll
    for (int j = 0; j < 4; ++j)
      #pragma unroll
      for (int r = 0; r < 8; ++r) {
        int grow = rowbase + rowOff + 16 * i + r + 8 * hi;
        int gcol = colbase + colOff + 16 * j + lo;
        size_t o = (size_t)grow * NTOT + gcol;
        W[o] -= acc[i][j][r];
      }
}

// ---------------- logdet from diag(L) ----------------

__device__ __forceinline__ void block_reduce_add(float v, float* target) {
  __shared__ float red[256];
  red[threadIdx.x] = v;
  __syncthreads();
  for (int s = blockDim.x >> 1; s > 0; s >>= 1) {
    if (threadIdx.x < s) red[threadIdx.x] += red[threadIdx.x + s];
    __syncthreads();
  }
  if (threadIdx.x == 0) atomicAdd(target, red[0]);
}

__global__ void k_logdet_diag(const float* __restrict__ W, float* target, int n) {
  float s = 0.0f;
  for (int i = blockIdx.x * blockDim.x + threadIdx.x; i < n; i += gridDim.x * blockDim.x)
    s += 2.0f * logf(W[(size_t)i * NTOT + i]);
  block_reduce_add(s, target);
}

// ---------------- triangular solves, x resident in LDS ----------------

__global__ void k_trsv_fwd(const float* __restrict__ L, float* __restrict__ x, int n) {
  extern __shared__ float xs[];
  for (int i = threadIdx.x; i < n; i += blockDim.x) xs[i] = x[i];
  __syncthreads();
  for (int j = 0; j < n; ++j) {
    if (threadIdx.x == 0) xs[j] /= L[(size_t)j * NTOT + j];
    __syncthreads();
    float v = xs[j];
    for (int i = j + 1 + threadIdx.x; i < n; i += blockDim.x)
      xs[i] -= L[(size_t)i * NTOT + j] * v;              // column j of L
    __syncthreads();
  }
  for (int i = threadIdx.x; i < n; i += blockDim.x) x[i] = xs[i];
}

__global__ void k_trsv_bwd(const float* __restrict__ L, float* __restrict__ x, int n) {
  extern __shared__ float xs[];
  for (int i = threadIdx.x; i < n; i += blockDim.x) xs[i] = x[i];
  __syncthreads();
  for (int j = n - 1; j >= 0; --j) {
    if (threadIdx.x == 0) xs[j] /= L[(size_t)j * NTOT + j];
    __syncthreads();
    float v = xs[j];
    for (int i = threadIdx.x; i < j; i += blockDim.x)
      xs[i] -= L[(size_t)j * NTOT + i] * v;              // row j of L (contiguous)
    __syncthreads();
  }
  for (int i = threadIdx.x; i < n; i += blockDim.x) x[i] = xs[i];
}

// ---------------- dot product -> atomic accumulate ----------------

__global__ void k_dot(const float* __restrict__ a, const float* __restrict__ b,
                      int n, float* target) {
  float s = 0.0f;
  for (int i = blockIdx.x * blockDim.x + threadIdx.x; i < n; i += gridDim.x * blockDim.x)
    s += a[i] * b[i];
  block_reduce_add(s, target);
}

// ---------------- 160x160 diagonal-block Cholesky logdets ----------------

__global__ void k_blkchol(const float* __restrict__ Kl, float* __restrict__ blk) {
  extern __shared__ float Ls[];               // DBLK x (DBLK+1)
  const int n = DBLK, LD = DBLK + 1;
  const int b = blockIdx.x, tid = threadIdx.x;   // DBLK threads
  const size_t base = ((size_t)b * DBLK) * NTOT + (size_t)b * DBLK;
  for (int c = 0; c < n; ++c)
    Ls[tid * LD + c] = Kl[base + (size_t)tid * NTOT + c];
  __syncthreads();
  for (int k = 0; k < n; ++k) {
    if (tid == k) Ls[k * LD + k] = sqrtf(Ls[k * LD + k]);
    __syncthreads();
    if (tid > k) Ls[tid * LD + k] /= Ls[k * LD + k];
    __syncthreads();
    if (tid > k) {
      float lik = Ls[tid * LD + k];
      for (int j = k + 1; j <= tid; ++j)
        Ls[tid * LD + j] -= lik * Ls[j * LD + k];
    }
    __syncthreads();
  }
  __shared__ float red[DBLK];
  red[tid] = 2.0f * logf(Ls[tid * LD + tid]);
  __syncthreads();
  if (tid == 0) {
    float s = 0.0f;
    for (int i = 0; i < n; ++i) s += red[i];
    blk[b] = s;
  }
}

// ---------------- BCE-with-logits sum ----------------

__global__ void k_bce(const float* __restrict__ l, const float* __restrict__ v,
                      int n, float* target) {
  float s = 0.0f;
  for (int i = blockIdx.x * blockDim.x + threadIdx.x; i < n; i += gridDim.x * blockDim.x) {
    float x = l[i], y = v[i];
    s += fmaxf(x, 0.0f) - x * y + log1pf(expf(-fabsf(x)));
  }
  block_reduce_add(s, target);
}

// ---------------- finalize ----------------

__global__ void k_finalize(const float* __restrict__ scal, const float* __restrict__ blk,
                           float* __restrict__ out) {
  if (threadIdx.x == 0 && blockIdx.x == 0) {
    float lml = -(1.0f / (2.0f * BSZ)) * scal[1] - 0.5f * scal[0]
                - 0.5f * (float)TMAXC * LOG_2PI;
    float t2 = 0.0f;
    for (int b = 0; b < BSZ - 1; ++b) t2 += blk[b];
    float gce = 0.5f * (float)NTOT * LOG_2PI + 0.5f * t2
                + 0.5f * scal[3] - 0.5f * scal[4] + 0.5f * scal[5];
    out[0] = lml;
    out[1] = gce;
    out[2] = scal[6] / (float)BSZ;
  }
}

// ---------------- orchestration ----------------

static void run_cholesky(float* W, hipStream_t stream) {
  const size_t potrf_lds = (size_t)NB * (NB + 1) * sizeof(float);
  const size_t trsm_lds  = 2 * potrf_lds;
  const size_t syrk_lds  = (size_t)2 * 128 * LDK * sizeof(float);
  for (int p = 0; p < NTOT / NB; ++p) {
    const int pcol = p * NB;
    const int s = pcol + NB;
    k_potrf<<<1, NB, potrf_lds, stream>>>(W, pcol);
    const int nbelow = (NTOT - s) / NB;
    if (nbelow > 0) {
      k_trsm<<<nbelow, NB, trsm_lds, stream>>>(W, pcol, s);
      const int nt = (NTOT - s) / 128;
      k_syrk<<<dim3(nt, nt), 256, syrk_lds, stream>>>(W, pcol, s);
    }
  }
}

extern "C" void kernel_launch(void* const* d_in, const int* in_sizes, int n_in,
                              void* d_out, int out_size, void* d_ws, size_t ws_size,
                              hipStream_t stream) {
  const float* mu_p   = (const float*)d_in[0];
  const float* K_p    = (const float*)d_in[1];
  const float* mu_l   = (const float*)d_in[2];
  const float* K_l    = (const float*)d_in[3];
  const float* mu_q   = (const float*)d_in[4];
  const float* K_q    = (const float*)d_in[5];
  const float* logits = (const float*)d_in[6];
  const float* vid    = (const float*)d_in[7];
  (void)in_sizes; (void)n_in; (void)out_size; (void)ws_size;

  const size_t NN = (size_t)NTOT * NTOT;
  float* W    = (float*)d_ws;                 // N x N factorization workspace
  float* v_d  = W + NN;                       // d = mu_p - mu_l
  float* v_x  = v_d + NTOT;                   // solve vector
  float* v_dK = v_x + NTOT;                   // diag(K_q)
  float* v_t  = v_dK + NTOT;                  // second solve vector
  float* scal = v_t + NTOT;                   // 64 scalar accumulators
  float* blk  = scal + 64;                    // 32 block log-dets
  float* out  = (float*)d_out;

  const size_t trsv_lds = (size_t)NTOT * sizeof(float);
  const size_t blk_lds  = (size_t)DBLK * (DBLK + 1) * sizeof(float);

  // accumulators must be zeroed every call (harness does not re-poison)
  k_zero<<<1, 128, 0, stream>>>(scal, 96);

  // ---- stage 1: lml via chol(K_p + K_l)
  k_add4<<<1024, 256, 0, stream>>>((const float4*)K_p, (const float4*)K_l,
                                   (float4*)W, NN / 4);
  k_vsub<<<(NTOT + 255) / 256, 256, 0, stream>>>(mu_p, mu_l, v_d, NTOT);
  k_diag<<<(NTOT + 255) / 256, 256, 0, stream>>>(K_q, v_dK, NTOT);

  run_cholesky(W, stream);
  k_logdet_diag<<<40, 256, 0, stream>>>(W, &scal[0], NTOT);
  hipMemcpyAsync(v_x, v_d, NTOT * sizeof(float), hipMemcpyDeviceToDevice, stream);
  k_trsv_fwd<<<1, 1024, trsv_lds, stream>>>(W, v_x, NTOT);
  k_trsv_bwd<<<1, 1024, trsv_lds, stream>>>(W, v_x, NTOT);
  k_dot<<<40, 256, 0, stream>>>(v_d, v_x, NTOT, &scal[1]);

  // ---- stage 2: gce via chol(K_l)
  hipMemcpyAsync(W, K_l, NN * sizeof(float), hipMemcpyDeviceToDevice, stream);
  run_cholesky(W, stream);

  hipMemcpyAsync(v_t, v_dK, NTOT * sizeof(float), hipMemcpyDeviceToDevice, stream);
  k_trsv_fwd<<<1, 1024, trsv_lds, stream>>>(W, v_t, NTOT);
  k_trsv_bwd<<<1, 1024, trsv_lds, stream>>>(W, v_t, NTOT);
  k_dot<<<40, 256, 0, stream>>>(v_dK, v_t, NTOT, &scal[3]);     // dK . K_l^{-1} dK

  hipMemcpyAsync(v_x, mu_l, NTOT * sizeof(float), hipMemcpyDeviceToDevice, stream);
  k_trsv_fwd<<<1, 1024, trsv_lds, stream>>>(W, v_x, NTOT);
  k_trsv_bwd<<<1, 1024, trsv_lds, stream>>>(W, v_x, NTOT);
  k_dot<<<40, 256, 0, stream>>>(mu_q, v_x, NTOT, &scal[4]);     // mu_q . inv_mu_l
  k_dot<<<40, 256, 0, stream>>>(mu_l, v_x, NTOT, &scal[5]);     // mu_l . inv_mu_l

  // ---- stage 3: block log-dets of K_l diagonal blocks (independent of factor)
  k_blkchol<<<BSZ, DBLK, blk_lds, stream>>>(K_l, blk);

  // ---- stage 4: reconstruction BCE
  k_bce<<<256, 256, 0, stream>>>(logits, vid, BSZ * TMAXC * PIXC, &scal[6]);

  // ---- combine
  k_finalize<<<1, 1, 0, stream>>>(scal, blk, out);
}